// KnowledgeDistillationKLDivLoss_ls_4707284156572
// MI455X (gfx1250) — compile-verified
//
#include <hip/hip_runtime.h>
#include <math.h>

// ---------------- problem constants ----------------
#define N_ROWS   1048576
#define BINS     17
#define NGROUPS  (N_ROWS / 4)     // 262144
#define CONVN    (2 * BINS - 1)   // 33
#define TEMP_F   10.0f
#define EPS_STD_F   1e-7f
#define EPS_RATIO_F 1e-8f

#define BLK_A    256              // 8 waves/block
#define ROWS_PER_WAVE 16
#define ROWS_PER_BLK  (8 * ROWS_PER_WAVE)       // 128
#define NBLK_A   (N_ROWS / ROWS_PER_BLK)        // 8192

typedef __attribute__((ext_vector_type(2))) float v2f;
typedef __attribute__((ext_vector_type(8))) float v8f;

// Row-sum of a 16x17 tile (zero-padded to 16x20) via V_WMMA_F32_16X16X4_F32.
// Data is placed in the B operand (K = column, N = row index = lane%16) and
// A = all-ones, so D[m][n] = sum_k B[k][n] : every lane gets its OWN row's
// sum (both lane halves jointly supply K, so the sum covers all columns).
// B layout (f32 4x16): vgpr v, lane l  ->  K = v + 2*(l>>4), N = l & 15.
__device__ __forceinline__ float rowsum_wmma(const float b[5][2]) {
    v2f ones; ones[0] = 1.0f; ones[1] = 1.0f;
    v8f acc = {0.f, 0.f, 0.f, 0.f, 0.f, 0.f, 0.f, 0.f};
#pragma unroll
    for (int t = 0; t < 5; ++t) {
        v2f bb; bb[0] = b[t][0]; bb[1] = b[t][1];
        acc = __builtin_amdgcn_wmma_f32_16x16x4_f32(
            /*neg_a=*/false, ones, /*neg_b=*/false, bb,
            /*c_mod=*/(short)0, acc, /*reuse_a=*/false, /*reuse_b=*/false);
    }
    return acc[0];  // D[m][n] identical over m; lane owns row n = lane&15
}

// Per-row: normalize (mean, std ddof=1), scale by 1/T, log-softmax pieces.
// Lane l handles row (base + l%16); h = l>>4 selects which columns it holds:
// tile t, slot v  ->  column c = 4t + 2h + v  (c >= 17 is zero padding).
__device__ __forceinline__ void analyze_row(
    const float* __restrict__ mat, int row, int h,
    float u[5][2],      // normalized/T values (junk at invalid cols)
    float e[5][2],      // exp(u - max), 0 at invalid cols
    float& L,           // logsumexp of u
    float& sumE)        // sum of e
{
    float vals[5][2];
#pragma unroll
    for (int t = 0; t < 5; ++t)
#pragma unroll
        for (int v = 0; v < 2; ++v) {
            int c = 4 * t + 2 * h + v;
            vals[t][v] = (c < BINS) ? mat[(size_t)row * BINS + c] : 0.0f;
        }

    float sx = rowsum_wmma(vals);                 // WMMA row sum
    float mean = sx * (1.0f / (float)BINS);

    float dev2[5][2];
#pragma unroll
    for (int t = 0; t < 5; ++t)
#pragma unroll
        for (int v = 0; v < 2; ++v) {
            int c = 4 * t + 2 * h + v;
            float d = (c < BINS) ? (vals[t][v] - mean) : 0.0f;
            dev2[t][v] = d * d;
        }
    float ssd = rowsum_wmma(dev2);                // WMMA sum of squared devs
    float var = fmaxf(ssd * (1.0f / (float)(BINS - 1)), 0.0f);
    float inv = 1.0f / (TEMP_F * (EPS_STD_F + sqrtf(var)));

    float mx = -3.0e38f;
#pragma unroll
    for (int t = 0; t < 5; ++t)
#pragma unroll
        for (int v = 0; v < 2; ++v) {
            int c = 4 * t + 2 * h + v;
            u[t][v] = (vals[t][v] - mean) * inv;
            if (c < BINS) mx = fmaxf(mx, u[t][v]);
        }
    // the partner lane (l ^ 16) holds the same row's other columns
    mx = fmaxf(mx, __shfl_xor(mx, 16, 32));

#pragma unroll
    for (int t = 0; t < 5; ++t)
#pragma unroll
        for (int v = 0; v < 2; ++v) {
            int c = 4 * t + 2 * h + v;
            e[t][v] = (c < BINS) ? __expf(u[t][v] - mx) : 0.0f;
        }
    sumE = rowsum_wmma(e);                        // WMMA sum of exps
    L = mx + __logf(sumE);
}

// ---------------- kernel 1: KL partials (WMMA) ----------------
__global__ void __launch_bounds__(BLK_A)
kd_kl_partial(const float* __restrict__ pred,
              const float* __restrict__ soft,
              const float* __restrict__ weight,
              float* __restrict__ partK,
              float* __restrict__ partW)
{
    const int lane = threadIdx.x & 31;
    const int wv   = threadIdx.x >> 5;
    const int rowBase = (blockIdx.x * 8 + wv) * ROWS_PER_WAVE;
    const int row = rowBase + (lane & 15);
    const int h = lane >> 4;

    float su[5][2], se[5][2], sL, sSumE;
    analyze_row(pred, row, h, su, se, sL, sSumE);
    float tu[5][2], te[5][2], tL, tSumE;
    analyze_row(soft, row, h, tu, te, tL, tSumE);

    const float invTS = 1.0f / tSumE;
    float kterm[5][2];
#pragma unroll
    for (int t = 0; t < 5; ++t)
#pragma unroll
        for (int v = 0; v < 2; ++v) {
            float p = te[t][v] * invTS;           // exactly 0 at padding
            kterm[t][v] = p * ((tu[t][v] - tL) - (su[t][v] - sL));
        }
    float kl = rowsum_wmma(kterm);                // per-row KL

    // deterministic wave reduction (rows duplicated in upper half -> zero it)
    float kw = (lane < 16) ? kl : 0.0f;
    float ww = (lane < 16) ? weight[rowBase + lane] : 0.0f;
#pragma unroll
    for (int off = 16; off >= 1; off >>= 1) {
        kw += __shfl_down(kw, off, 32);
        ww += __shfl_down(ww, off, 32);
    }

    __shared__ float sK[8], sW[8];
    if (lane == 0) { sK[wv] = kw; sW[wv] = ww; }
    __syncthreads();
    if (threadIdx.x == 0) {
        float a = 0.0f, b = 0.0f;
#pragma unroll
        for (int i = 0; i < 8; ++i) { a += sK[i]; b += sW[i]; }
        partK[blockIdx.x] = a;
        partW[blockIdx.x] = b;
    }
}

// ---------------- kernel 2: final deterministic reduction ----------------
__global__ void __launch_bounds__(256)
kd_finalize(const float* __restrict__ partK,
            const float* __restrict__ partW,
            float* __restrict__ out)
{
    __shared__ float rK[256], rW[256];
    float a = 0.0f, b = 0.0f;
    for (int i = threadIdx.x; i < NBLK_A; i += 256) { a += partK[i]; b += partW[i]; }
    rK[threadIdx.x] = a; rW[threadIdx.x] = b;
    __syncthreads();
    for (int s = 128; s >= 1; s >>= 1) {
        if ((int)threadIdx.x < s) {
            rK[threadIdx.x] += rK[threadIdx.x + s];
            rW[threadIdx.x] += rW[threadIdx.x + s];
        }
        __syncthreads();
    }
    if (threadIdx.x == 0) {
        float loss_ls = (rK[0] / (float)N_ROWS) * (TEMP_F * TEMP_F);
        out[0] = loss_ls * (rW[0] / (float)N_ROWS);   // LOSS_WEIGHT == 1
    }
}

// ---------------- kernel 3: softmax + conv ratio ----------------
__device__ __forceinline__ void softmax_row17(const float* __restrict__ p, float o[BINS]) {
    float m = p[0];
#pragma unroll
    for (int i = 1; i < BINS; ++i) m = fmaxf(m, p[i]);
    float s = 0.0f;
#pragma unroll
    for (int i = 0; i < BINS; ++i) { o[i] = __expf(p[i] - m); s += o[i]; }
    float inv = 1.0f / s;
#pragma unroll
    for (int i = 0; i < BINS; ++i) o[i] *= inv;
}

__device__ __forceinline__ void conv17(const float a[BINS], const float b[BINS], float c[CONVN]) {
#pragma unroll
    for (int k = 0; k < CONVN; ++k) c[k] = 0.0f;
#pragma unroll
    for (int i = 0; i < BINS; ++i)
#pragma unroll
        for (int j = 0; j < BINS; ++j)
            c[i + j] = fmaf(a[i], b[j], c[i + j]);
}

__global__ void __launch_bounds__(256)
kd_conv_ratio(const float* __restrict__ soft, float* __restrict__ out)
{
    const int g = blockIdx.x * blockDim.x + threadIdx.x;
    if (g >= NGROUPS) return;
    const float* base = soft + (size_t)g * 4 * BINS;

    float a[BINS], b[BINS], ctb[CONVN], clr[CONVN];
    softmax_row17(base + 0 * BINS, a);   // top
    softmax_row17(base + 1 * BINS, b);   // bottom
    conv17(a, b, ctb);
    softmax_row17(base + 2 * BINS, a);   // left
    softmax_row17(base + 3 * BINS, b);   // right
    conv17(a, b, clr);

    float* o = out + 1 + (size_t)g * CONVN;
#pragma unroll
    for (int k = 0; k < CONVN; ++k)
        o[k] = clr[k] / (ctb[k] + EPS_RATIO_F);
}

// ---------------- launch ----------------
extern "C" void kernel_launch(void* const* d_in, const int* in_sizes, int n_in,
                              void* d_out, int out_size, void* d_ws, size_t ws_size,
                              hipStream_t stream) {
    (void)in_sizes; (void)n_in; (void)out_size; (void)ws_size;
    const float* pred   = (const float*)d_in[0];
    const float* soft   = (const float*)d_in[1];
    const float* weight = (const float*)d_in[2];
    float* out = (float*)d_out;

    float* partK = (float*)d_ws;           // NBLK_A floats
    float* partW = partK + NBLK_A;         // NBLK_A floats  (64 KB total)

    kd_kl_partial<<<NBLK_A, BLK_A, 0, stream>>>(pred, soft, weight, partK, partW);
    kd_finalize<<<1, 256, 0, stream>>>(partK, partW, out);
    kd_conv_ratio<<<NGROUPS / 256, 256, 0, stream>>>(soft, out);
}